// SLatMeshDecoder_18743237280119
// MI455X (gfx1250) — compile-verified
//
#include <hip/hip_runtime.h>
#include <hip/hip_bf16.h>

#define RES 128
#define RES_V 129

constexpr int   kNVerts       = RES_V * RES_V * RES_V;   // 2,146,689
constexpr int   kNCells       = RES * RES * RES;         // 2,097,152
constexpr float kSdfBias      = -1.0f / (float)RES;
constexpr float kDeformScale  = (float)((1.0 - 1e-8) / (2.0 * (double)RES));
constexpr int   kVoxPerBlock  = 128;   // 128 * 404B = 51,712B LDS tile

// ---- CDNA5 async global->LDS path (guarded so compile never breaks) -------
#if defined(__HIP_DEVICE_COMPILE__) && defined(__gfx1250__) && \
    __has_builtin(__builtin_amdgcn_global_load_async_to_lds_b128) && \
    __has_builtin(__builtin_amdgcn_global_load_async_to_lds_b32)
#define USE_ASYNC_LDS 1
#else
#define USE_ASYNC_LDS 0
#endif

#if USE_ASYNC_LDS
typedef int v4i __attribute__((ext_vector_type(4)));
typedef __attribute__((address_space(1))) v4i* gv4p;   // global v4i*
typedef __attribute__((address_space(3))) v4i* lv4p;   // LDS v4i*
typedef __attribute__((address_space(1))) int* gi32p;  // global int*
typedef __attribute__((address_space(3))) int* li32p;  // LDS int*

__device__ __forceinline__ void async_copy_b128(const float* g, float* l) {
  v4i* gg = reinterpret_cast<v4i*>(const_cast<float*>(g));
  v4i* ll = reinterpret_cast<v4i*>(l);
  __builtin_amdgcn_global_load_async_to_lds_b128((gv4p)gg, (lv4p)ll, 0, 0);
}
__device__ __forceinline__ void async_copy_b32(const float* g, float* l) {
  int* gg = reinterpret_cast<int*>(const_cast<float*>(g));
  int* ll = reinterpret_cast<int*>(l);
  __builtin_amdgcn_global_load_async_to_lds_b32((gi32p)gg, (li32p)ll, 0, 0);
}
__device__ __forceinline__ void wait_async0() {
#if __has_builtin(__builtin_amdgcn_s_wait_asynccnt)
  __builtin_amdgcn_s_wait_asynccnt(0);
#else
  asm volatile("s_wait_asynccnt 0" ::: "memory");
#endif
}
#endif  // USE_ASYNC_LDS

__device__ __forceinline__ void atomAddF(float* p, float v) {
#if defined(__HIP_DEVICE_COMPILE__)
  unsafeAtomicAdd(p, v);   // -> global_atomic_add_f32 (no CAS loop)
#else
  atomicAdd(p, v);
#endif
}

// ---------------------------------------------------------------------------
// Scatter: per-voxel values -> vertex accumulator (stride 12: 10 attrs, cnt),
// plus owner grid for the weights scatter (max voxel index == last-write-wins).
// ---------------------------------------------------------------------------
__global__ void __launch_bounds__(kVoxPerBlock)
slat_scatter_kernel(const int* __restrict__ coords, const float* __restrict__ feats,
                    float* __restrict__ sums, int* __restrict__ owner, int n)
{
  __shared__ __align__(16) float sfeat[kVoxPerBlock * 101];

  const int vbase = blockIdx.x * kVoxPerBlock;
  int count = n - vbase;
  if (count <= 0) return;
  if (count > kVoxPerBlock) count = kVoxPerBlock;

  // Stage this block's contiguous feats tile into LDS.
  const float* gsrc = feats + (size_t)vbase * 101;  // block base is 16B aligned
  const int nfl = count * 101;
  const int n4  = nfl >> 2;
#if USE_ASYNC_LDS
  for (int i = threadIdx.x; i < n4; i += blockDim.x)
    async_copy_b128(gsrc + 4 * i, sfeat + 4 * i);
  for (int i = (n4 << 2) + threadIdx.x; i < nfl; i += blockDim.x)
    async_copy_b32(gsrc + i, sfeat + i);
  wait_async0();
#else
  {
    const float4* g4 = (const float4*)gsrc;
    float4*       s4 = (float4*)sfeat;
    for (int i = threadIdx.x; i < n4; i += blockDim.x) s4[i] = g4[i];
    for (int i = (n4 << 2) + threadIdx.x; i < nfl; i += blockDim.x) sfeat[i] = gsrc[i];
  }
#endif
  __syncthreads();

  const int t = threadIdx.x;
  if (t >= count) return;
  const int v  = vbase + t;
  const int cx = coords[3 * (size_t)v + 0];
  const int cy = coords[3 * (size_t)v + 1];
  const int cz = coords[3 * (size_t)v + 2];
  const float* f = sfeat + t * 101;   // stride 101 dwords: conflict-free across lanes

  atomicMax(&owner[(cx * RES + cy) * RES + cz], v);

  #pragma unroll
  for (int c = 0; c < 8; ++c) {
    const int dx = c & 1, dy = (c >> 1) & 1, dz = (c >> 2) & 1;
    const int vid = ((cx + dx) * RES_V + (cy + dy)) * RES_V + (cz + dz);
    float* s = sums + (size_t)vid * 12;
    atomAddF(s + 0, f[c] + kSdfBias);          // sdf
    atomAddF(s + 1, f[8 + 3 * c + 0]);         // deform
    atomAddF(s + 2, f[8 + 3 * c + 1]);
    atomAddF(s + 3, f[8 + 3 * c + 2]);
    #pragma unroll
    for (int j = 0; j < 6; ++j)
      atomAddF(s + 4 + j, f[53 + 6 * c + j]);  // color
    atomAddF(s + 10, 1.0f);                    // count
  }
}

// ---------------------------------------------------------------------------
// Finalize: per-vertex mean, tanh deform -> position, write [pos, sdf, color].
// ---------------------------------------------------------------------------
__global__ void __launch_bounds__(256)
slat_finalize_kernel(const float* __restrict__ sums, float* __restrict__ vout)
{
  const int v = blockIdx.x * blockDim.x + threadIdx.x;
  if (v >= kNVerts) return;
  const float4* row = (const float4*)(sums + (size_t)v * 12);  // 48B rows: 16B aligned
  const float4 a = row[0];   // sdf, deform xyz
  const float4 b = row[1];   // color 0..3
  const float4 c = row[2];   // color 4,5, cnt, pad
  const float cd = fmaxf(c.z, 1.0f);

  const int x   = v / (RES_V * RES_V);
  const int rem = v - x * (RES_V * RES_V);
  const int y   = rem / RES_V;
  const int z   = rem - y * RES_V;

  const float inv_res = 1.0f / (float)RES;
  const float px = (float)x * inv_res - 0.5f + kDeformScale * tanhf(a.y / cd);
  const float py = (float)y * inv_res - 0.5f + kDeformScale * tanhf(a.z / cd);
  const float pz = (float)z * inv_res - 0.5f + kDeformScale * tanhf(a.w / cd);

  float* o = vout + (size_t)v * 10;
  o[0] = px;       o[1] = py;       o[2] = pz;
  o[3] = a.x / cd;
  o[4] = b.x / cd; o[5] = b.y / cd; o[6] = b.z / cd; o[7] = b.w / cd;
  o[8] = c.x / cd; o[9] = c.y / cd;
}

// ---------------------------------------------------------------------------
// Weights: dense 128^3 x 21 grid; owner>=0 -> copy feats[32:53], else zeros.
// Fuses zero-init with scatter (each output byte written exactly once).
// ---------------------------------------------------------------------------
__global__ void __launch_bounds__(256)
slat_weights_kernel(const float* __restrict__ feats, const int* __restrict__ owner,
                    float* __restrict__ wout)
{
  const int cell = blockIdx.x * blockDim.x + threadIdx.x;
  if (cell >= kNCells) return;
  const int own = owner[cell];
  float* o = wout + (size_t)cell * 21;
  if (own >= 0) {
    const float* f = feats + (size_t)own * 101 + 32;
    #pragma unroll
    for (int j = 0; j < 21; ++j) o[j] = f[j];
  } else {
    #pragma unroll
    for (int j = 0; j < 21; ++j) o[j] = 0.0f;
  }
}

// ---------------------------------------------------------------------------
extern "C" void kernel_launch(void* const* d_in, const int* in_sizes, int n_in,
                              void* d_out, int out_size, void* d_ws, size_t ws_size,
                              hipStream_t stream)
{
  const int*   coords = (const int*)d_in[0];
  const float* feats  = (const float*)d_in[1];
  const int    n      = in_sizes[0] / 3;          // 400,000 voxels

  // Workspace: vertex accumulator (stride 12 floats) + owner grid.
  float* sums = (float*)d_ws;
  const size_t sums_bytes = (size_t)kNVerts * 12 * sizeof(float);   // ~103 MB
  int* owner = (int*)((char*)d_ws + sums_bytes);
  const size_t owner_bytes = (size_t)kNCells * sizeof(int);         // 8 MB

  // Output: v_out (kNVerts*10) then weights_d (kNCells*21), concatenated flat.
  float* vout = (float*)d_out;
  float* wout = vout + (size_t)kNVerts * 10;

  (void)hipMemsetAsync(sums, 0, sums_bytes, stream);          // zero accumulator
  (void)hipMemsetAsync(owner, 0xFF, owner_bytes, stream);     // owner = -1

  const int sblocks = (n + kVoxPerBlock - 1) / kVoxPerBlock;
  hipLaunchKernelGGL(slat_scatter_kernel, dim3(sblocks), dim3(kVoxPerBlock), 0, stream,
                     coords, feats, sums, owner, n);

  const int fblocks = (kNVerts + 255) / 256;
  hipLaunchKernelGGL(slat_finalize_kernel, dim3(fblocks), dim3(256), 0, stream,
                     sums, vout);

  const int wblocks = (kNCells + 255) / 256;
  hipLaunchKernelGGL(slat_weights_kernel, dim3(wblocks), dim3(256), 0, stream,
                     feats, owner, wout);
}